// DeepGP_19009525252313
// MI455X (gfx1250) — compile-verified
//
#include <hip/hip_runtime.h>
#include <math.h>

typedef __attribute__((ext_vector_type(2))) float v2f;
typedef __attribute__((ext_vector_type(8))) float v8f;

#define NN 256
#define PP 33

// ---------------------------------------------------------------------------
// K1: h[i,p] = sigmoid( sum_k W2[k]*tanh(W1[k]*t + b1[k]) + b2 ),
//     t = a + (b-a)*p/32 ;  wq[i,p] = simpson_w[p]/3 * (b-a)/32
// grid = 256 blocks (one per i), block = 64 threads (p = tid < 33)
// ---------------------------------------------------------------------------
__global__ void dgp_h_wq(const float* __restrict__ x_train,
                         const float* __restrict__ W1, const float* __restrict__ b1,
                         const float* __restrict__ W2, const float* __restrict__ b2,
                         float* __restrict__ h, float* __restrict__ wq) {
    const int i = blockIdx.x;
    const int p = threadIdx.x;
    if (p >= PP) return;
    const float a  = x_train[2 * i + 0];
    const float bb = x_train[2 * i + 1];
    const float t  = a + (bb - a) * ((float)p * (1.0f / 32.0f));
    float z = 0.0f;
#pragma unroll
    for (int k = 0; k < 10; ++k)
        z += W2[k] * tanhf(W1[k] * t + b1[k]);
    z += b2[0];
    h[i * PP + p] = 1.0f / (1.0f + __expf(-z));
    float w = (p == 0 || p == 32) ? 1.0f : ((p & 1) ? 4.0f : 2.0f);
    wq[i * PP + p] = (w * (1.0f / 3.0f)) * (bb - a) * (1.0f / 32.0f);
}

// ---------------------------------------------------------------------------
// K2: kyy[i,j] = sf^2 * sum_p wq[i,p] * sum_q wq[j,q]*exp(-0.5(h_ip-h_jq)^2/l^2)
//     one wave32 per (i,j); lanes own q (lane 0 also owns q=32)
// grid = 8192 blocks * 256 threads = 65536 waves... (8 waves/block)
// ---------------------------------------------------------------------------
__global__ void dgp_kyy(const float* __restrict__ h, const float* __restrict__ wq,
                        float* __restrict__ kyy,
                        const float* __restrict__ sf, const float* __restrict__ ls,
                        const float* __restrict__ sn) {
    const int wid  = blockIdx.x * 8 + (threadIdx.x >> 5);
    const int lane = threadIdx.x & 31;
    const int i = wid >> 8;
    const int j = wid & 255;
    const float l = ls[0];
    const float c = -0.5f / (l * l);

    const float hj0 = h[j * PP + lane];
    const float wj0 = wq[j * PP + lane];
    const float hj1 = (lane == 0) ? h[j * PP + 32] : 0.0f;
    const float wj1 = (lane == 0) ? wq[j * PP + 32] : 0.0f;

    float acc = 0.0f;
    for (int p = 0; p < PP; ++p) {
        const float hp = h[i * PP + p];
        const float wp = wq[i * PP + p];
        float d0 = hp - hj0;
        float e  = wj0 * __expf(c * d0 * d0);
        if (lane == 0) {
            float d1 = hp - hj1;
            e += wj1 * __expf(c * d1 * d1);
        }
        acc += wp * e;
    }
#pragma unroll
    for (int off = 16; off > 0; off >>= 1)
        acc += __shfl_down(acc, off, 32);
    if (lane == 0) {
        float v = sf[0] * sf[0] * acc;
        if (i == j) v += sn[0] * sn[0];
        kyy[i * NN + j] = v;
    }
}

// ---------------------------------------------------------------------------
// K3: min eigenvalue estimate via power iteration (lamMax) then shifted
//     power iteration on (lamMax*I - K). Single workgroup, 256 threads.
// ---------------------------------------------------------------------------
__global__ void dgp_mineig(const float* __restrict__ K, float* __restrict__ mineOut) {
    __shared__ float x[NN];
    __shared__ float red[NN];
    const int tid = threadIdx.x;
    const float* row = K + tid * NN;

    x[tid] = 1.0f;
    __syncthreads();
    float lamMax = 0.0f;
    for (int it = 0; it < 64; ++it) {
        float y = 0.0f;
        for (int jj = 0; jj < NN; ++jj) y += row[jj] * x[jj];
        red[tid] = y * y;
        __syncthreads();
        for (int s = 128; s > 0; s >>= 1) {
            if (tid < s) red[tid] += red[tid + s];
            __syncthreads();
        }
        const float nrm = sqrtf(red[0]);
        lamMax = nrm;
        x[tid] = y / fmaxf(nrm, 1e-30f);
        __syncthreads();
    }
    x[tid] = (tid & 1) ? -1.0f : 1.0f;
    __syncthreads();
    float lamB = 0.0f;
    for (int it = 0; it < 96; ++it) {
        float y = 0.0f;
        for (int jj = 0; jj < NN; ++jj) y += row[jj] * x[jj];
        y = lamMax * x[tid] - y;          // shifted operator
        red[tid] = y * y;
        __syncthreads();
        for (int s = 128; s > 0; s >>= 1) {
            if (tid < s) red[tid] += red[tid + s];
            __syncthreads();
        }
        const float nrm = sqrtf(red[0]);
        lamB = nrm;
        x[tid] = y / fmaxf(nrm, 1e-30f);
        __syncthreads();
    }
    if (tid == 0) mineOut[0] = lamMax - lamB;
}

// ---------------------------------------------------------------------------
// K4: conditional jitter: if mine<1e-6, kyy += 1.1*|1e-6 - eye*mine|
// ---------------------------------------------------------------------------
__global__ void dgp_jitter(float* __restrict__ kyy, const float* __restrict__ mine) {
    const float m = mine[0];
    if (m >= 1e-6f) return;
    const int i = blockIdx.x, j = threadIdx.x;
    const float d = (i == j) ? m : 0.0f;
    kyy[i * NN + j] += 1.1f * fabsf(1e-6f - d);
}

// ---------------------------------------------------------------------------
// K5: in-place blocked Cholesky (lower), NB=16, trailing SYRK update done
//     with V_WMMA_F32_16X16X4_F32 (4 chained WMMAs per 16x16x16 tile).
//     Single workgroup, 256 threads = 8 waves. Full-square trailing update
//     keeps the trailing matrix symmetric (no mirror bookkeeping).
// ---------------------------------------------------------------------------
__global__ void dgp_cholesky(float* __restrict__ A) {
    __shared__ float Lkk[16][17];
    __shared__ float P[NN][16];           // panel rows (absolute row index)
    const int tid  = threadIdx.x;
    const int lane = tid & 31;
    const int wave = tid >> 5;
    const int m    = lane & 15;           // M (or N) index within tile
    const int half = lane >> 4;           // 0: lanes 0-15, 1: lanes 16-31

    for (int kb = 0; kb < 16; ++kb) {
        const int k0 = kb * 16;
        // --- load diagonal block ---
        {
            const int r = tid >> 4, cc = tid & 15;
            Lkk[r][cc] = A[(k0 + r) * NN + k0 + cc];
        }
        __syncthreads();
        // --- factor 16x16 diagonal block (thread 0, serial) ---
        if (tid == 0) {
            for (int c = 0; c < 16; ++c) {
                const float d = sqrtf(fmaxf(Lkk[c][c], 1e-20f));
                Lkk[c][c] = d;
                const float inv = 1.0f / d;
                for (int r = c + 1; r < 16; ++r) Lkk[r][c] *= inv;
                for (int c2 = c + 1; c2 < 16; ++c2)
                    for (int r = c2; r < 16; ++r)
                        Lkk[r][c2] -= Lkk[r][c] * Lkk[c2][c];
            }
        }
        __syncthreads();
        // --- write back diag block (lower part) ---
        {
            const int r = tid >> 4, cc = tid & 15;
            if (cc <= r) A[(k0 + r) * NN + k0 + cc] = Lkk[r][cc];
            else         A[(k0 + r) * NN + k0 + cc] = 0.0f;
        }
        // --- panel triangular solve: rows below diag block ---
        const int nrows = NN - k0 - 16;
        if (tid < nrows) {
            const int row = k0 + 16 + tid;
            float xr[16];
#pragma unroll
            for (int c = 0; c < 16; ++c) xr[c] = A[row * NN + k0 + c];
            for (int c = 0; c < 16; ++c) {
                float s = xr[c];
                for (int d = 0; d < c; ++d) s -= xr[d] * Lkk[c][d];
                xr[c] = s / Lkk[c][c];
            }
#pragma unroll
            for (int c = 0; c < 16; ++c) {
                A[row * NN + k0 + c] = xr[c];
                P[row][c] = xr[c];
            }
        }
        __syncthreads();
        // --- trailing update via WMMA: A[ib,jb] -= P[ib] * P[jb]^T ---
        const int nt = 15 - kb;
        const int ntiles = nt * nt;
        for (int t = wave; t < ntiles; t += 8) {
            const int ib = kb + 1 + t / nt;
            const int jb = kb + 1 + t % nt;
            v8f acc;
#pragma unroll
            for (int r = 0; r < 8; ++r)
                acc[r] = A[(ib * 16 + r + 8 * half) * NN + jb * 16 + m];
#pragma unroll
            for (int kk = 0; kk < 16; kk += 4) {
                v2f av, bv;
                // A fragment (negated)
                av.x = -P[ib * 16 + m][kk + 2 * half];
                av.y = -P[ib * 16 + m][kk + 2 * half + 1];
                // B fragment = P[jb]^T : lane owns column n=m of B = row n of P[jb]
                bv.x =  P[jb * 16 + m][kk + 2 * half];
                bv.y =  P[jb * 16 + m][kk + 2 * half + 1];
                acc = __builtin_amdgcn_wmma_f32_16x16x4_f32(
                        /*neg_a=*/false, av, /*neg_b=*/false, bv,
                        /*c_mod=*/(short)0, acc,
                        /*reuse_a=*/false, /*reuse_b=*/false);
            }
#pragma unroll
            for (int r = 0; r < 8; ++r)
                A[(ib * 16 + r + 8 * half) * NN + jb * 16 + m] = acc[r];
        }
        __syncthreads();
    }
}

// ---------------------------------------------------------------------------
// K6: solve (L L^T) v = y via forward + backward substitution.
//     Single workgroup, 256 threads, one row per thread.
// ---------------------------------------------------------------------------
__global__ void dgp_solve(const float* __restrict__ L, const float* __restrict__ y,
                          float* __restrict__ vout) {
    __shared__ float z[NN];
    __shared__ float acc[NN];
    const int tid = threadIdx.x;

    acc[tid] = 0.0f;
    __syncthreads();
    for (int k = 0; k < NN; ++k) {
        if (tid == k) z[k] = (y[k] - acc[k]) / L[k * NN + k];
        __syncthreads();
        if (tid > k) acc[tid] += L[tid * NN + k] * z[k];
        __syncthreads();
    }
    acc[tid] = 0.0f;
    __syncthreads();
    for (int k = NN - 1; k >= 0; --k) {
        if (tid == k) z[k] = (z[k] - acc[k]) / L[k * NN + k];  // reuse z as v
        __syncthreads();
        if (tid < k) acc[tid] += L[k * NN + tid] * z[k];
        __syncthreads();
    }
    vout[tid] = z[tid];
}

// ---------------------------------------------------------------------------
// K7: c = L^T (upper triangular), zeros below diagonal
// ---------------------------------------------------------------------------
__global__ void dgp_transpose(const float* __restrict__ L, float* __restrict__ c) {
    const int i = blockIdx.x, j = threadIdx.x;
    c[i * NN + j] = (j >= i) ? L[j * NN + i] : 0.0f;
}

// ---------------------------------------------------------------------------
extern "C" void kernel_launch(void* const* d_in, const int* in_sizes, int n_in,
                              void* d_out, int out_size, void* d_ws, size_t ws_size,
                              hipStream_t stream) {
    const float* x_train = (const float*)d_in[0];   // (256,2)
    const float* y_train = (const float*)d_in[1];   // (256,)
    const float* W1      = (const float*)d_in[2];   // (10,1)
    const float* b1      = (const float*)d_in[3];   // (10,)
    const float* W2      = (const float*)d_in[4];   // (1,10)
    const float* b2      = (const float*)d_in[5];   // (1,)
    const float* sf      = (const float*)d_in[6];
    const float* ls      = (const float*)d_in[7];
    const float* sn      = (const float*)d_in[8];

    float* ws   = (float*)d_ws;
    float* h    = ws;                       // 256*33
    float* wq   = ws + NN * PP;             // 256*33
    float* kyy  = ws + 2 * NN * PP;         // 256*256
    float* mine = kyy + NN * NN;            // 1

    float* c_out = (float*)d_out;           // 65536
    float* v_out = c_out + NN * NN;         // 256

    dgp_h_wq<<<NN, 64, 0, stream>>>(x_train, W1, b1, W2, b2, h, wq);
    dgp_kyy<<<(NN * NN) / 8, 256, 0, stream>>>(h, wq, kyy, sf, ls, sn);
    dgp_mineig<<<1, NN, 0, stream>>>(kyy, mine);
    dgp_jitter<<<NN, NN, 0, stream>>>(kyy, mine);
    dgp_cholesky<<<1, 256, 0, stream>>>(kyy);
    dgp_solve<<<1, NN, 0, stream>>>(kyy, y_train, v_out);
    dgp_transpose<<<NN, NN, 0, stream>>>(kyy, c_out);
}